// GenGenerator_81509889343846
// MI455X (gfx1250) — compile-verified
//
#include <hip/hip_runtime.h>
#include <math.h>

typedef __attribute__((ext_vector_type(2))) float v2f;
typedef __attribute__((ext_vector_type(8))) float v8f;

#define B_SZ   8192
#define NCOL   32
#define D_SZ   512
#define CAT    10
#define NPAIR  16

// One block = 4 wave32 = 128 threads; owns 16 batch rows.
// Each wave handles 4 column-pairs; per pair: two f32 WMMA GEMM chains
// (cat logits 16x512 @ 512x16-padded, num dot 16x512 @ 512x16 col0-only).
__global__ __launch_bounds__(128, 1) void fused_tab_wmma(
    const float* __restrict__ x,     // (B, 32, 512)
    const float* __restrict__ Wnum,  // (16, 512, 1)
    const float* __restrict__ bnum,  // (16, 1)
    const float* __restrict__ Wcat,  // (16, 512, 10)
    const float* __restrict__ bcat,  // (16, 10)
    float* __restrict__ out)         // (B, 32)
{
  const int lane  = threadIdx.x & 31;
  const int wave  = threadIdx.x >> 5;     // 0..3
  const int row0  = blockIdx.x * 16;      // 16 batch rows per block
  const int m     = lane & 15;            // A-matrix row this lane feeds
  const int khalf = lane >> 4;            // 0/1 -> K offset 0/2 inside k-block
  const int col   = lane & 15;            // B/C column this lane holds
  const float maskC = (col < CAT) ? 1.0f : 0.0f;   // zero-pad cols 10..15
  const float maskN = (col == 0) ? 1.0f : 0.0f;    // num uses only col 0
  const int   colc  = (col < CAT) ? col : (CAT - 1);

  for (int p = 0; p < 4; ++p) {
    const int n = wave * 4 + p;           // pair index 0..15
    const float* Ac = x + ((size_t)(row0 + m) * NCOL + (2 * n + 1)) * D_SZ;
    const float* An = x + ((size_t)(row0 + m) * NCOL + (2 * n    )) * D_SZ;
    const float* Bc = Wcat + (size_t)n * D_SZ * CAT + colc;  // + d*CAT
    const float* Bn = Wnum + (size_t)n * D_SZ;               // + d (col 0)

    if (p < 3) {  // prefetch next pair's A rows (global_prefetch_b8)
      __builtin_prefetch(x + ((size_t)(row0 + m) * NCOL + (2 * (n + 1) + 1)) * D_SZ, 0, 0);
      __builtin_prefetch(x + ((size_t)(row0 + m) * NCOL + (2 * (n + 1)    )) * D_SZ, 0, 0);
    }

    v8f accC = {};
    v8f accN = {};
    #pragma unroll 4
    for (int k = 0; k < D_SZ; k += 4) {
      const int ka = k + 2 * khalf;
      // A fragments: lane m holds A[m][ka], A[m][ka+1] (b64 load)
      v2f a_c = { Ac[ka], Ac[ka + 1] };
      v2f a_n = { An[ka], An[ka + 1] };
      // B fragments: lane col holds B[ka][col], B[ka+1][col]; mask pad cols
      v2f b_c = { Bc[(size_t)ka * CAT] * maskC,
                  Bc[(size_t)(ka + 1) * CAT] * maskC };
      v2f b_n = { Bn[ka] * maskN, Bn[ka + 1] * maskN };
      accC = __builtin_amdgcn_wmma_f32_16x16x4_f32(
          false, a_c, false, b_c, (short)0, accC, false, false);
      accN = __builtin_amdgcn_wmma_f32_16x16x4_f32(
          false, a_n, false, b_n, (short)0, accN, false, false);
    }

    const float biasC = bcat[n * CAT + colc];
    const float biasN = bnum[n];

    #pragma unroll
    for (int r = 0; r < 8; ++r) {
      // ---- categorical: argmax over the 10 logits spread across 16 lanes
      float v  = (col < CAT) ? (accC[r] + biasC) : -__builtin_inff();
      int  idx = col;
      #pragma unroll
      for (int off = 8; off >= 1; off >>= 1) {
        float ov = __shfl_xor(v, off, 16);
        int   oi = __shfl_xor(idx, off, 16);
        if (ov > v || (ov == v && oi < idx)) { v = ov; idx = oi; }
      }
      // ---- numeric: lane with N==0 holds the dot product
      const float tn = tanhf(accN[r] + biasN);
      if (col == 0) {  // lanes 0 and 16 -> rows r and r+8
        const int row = row0 + r + 8 * khalf;
        out[(size_t)row * NCOL + 2 * n    ] = tn;
        out[(size_t)row * NCOL + 2 * n + 1] = (float)idx;
      }
    }
  }
}

extern "C" void kernel_launch(void* const* d_in, const int* in_sizes, int n_in,
                              void* d_out, int out_size, void* d_ws, size_t ws_size,
                              hipStream_t stream) {
  const float* x    = (const float*)d_in[0];
  const float* Wnum = (const float*)d_in[1];
  const float* bnum = (const float*)d_in[2];
  const float* Wcat = (const float*)d_in[3];
  const float* bcat = (const float*)d_in[4];
  float* out = (float*)d_out;

  dim3 grid(B_SZ / 16);   // 512 tiles of 16 batch rows
  dim3 block(128);        // 4 wave32
  fused_tab_wmma<<<grid, block, 0, stream>>>(x, Wnum, bnum, Wcat, bcat, out);
}